// MultiInputMLPUpdate_8461085573254
// MI455X (gfx1250) — compile-verified
//
#include <hip/hip_runtime.h>
#include <hip/hip_bf16.h>

// CDNA5 / gfx1250, wave32. fp32 WMMA (V_WMMA_F32_16X16X4_F32) path.

typedef float v2f __attribute__((ext_vector_type(2)));
typedef float v8f __attribute__((ext_vector_type(8)));

#define N_NODES 80000
#define N_EDGES 1280000

__device__ __forceinline__ v8f wmma4(v2f a, v2f b, v8f c) {
  // D = A(16x4) x B(4x16) + C, fp32
  return __builtin_amdgcn_wmma_f32_16x16x4_f32(false, a, false, b, (short)0, c,
                                               false, false);
}
__device__ __forceinline__ v2f ld2(const float* p) { return *(const v2f*)p; }
__device__ __forceinline__ v2f ld2_nt(const float* p) {
  // streamed data (read once): non-temporal hint, keep L2 for the p-table
  return __builtin_nontemporal_load((const v2f*)p);
}
__device__ __forceinline__ void lds_fence() {
  // drain DS counter + compiler memory barrier: makes per-wave LDS phases
  // (store -> cross-lane load) safe without a workgroup barrier.
  asm volatile("s_wait_dscnt 0" ::: "memory");
}

// ---------------------------------------------------------------------------
// Kernel 1: p[v] = h[v] @ W1^T + e[v] @ W2^T + (b1+b2)   [N_NODES x 64]
// One 16-node tile per wave. Weights staged in LDS with padded strides so
// B-fragment ds_load_b64 reads are bank-conflict free.
// ---------------------------------------------------------------------------
__global__ __launch_bounds__(128) void node_proj_kernel(
    const float* __restrict__ h, const float* __restrict__ e,
    const float* __restrict__ W1, const float* __restrict__ b1,
    const float* __restrict__ W2, const float* __restrict__ b2,
    float* __restrict__ p) {
  __shared__ float sW1[64 * 68];  // W1[n][k], stride 68
  __shared__ float sW2[64 * 36];  // W2[n][k], stride 36
  __shared__ float sB[64];

  const int t = threadIdx.x;
  for (int i = t; i < 64 * 64; i += 128) sW1[(i >> 6) * 68 + (i & 63)] = W1[i];
  for (int i = t; i < 64 * 32; i += 128) sW2[(i >> 5) * 36 + (i & 31)] = W2[i];
  if (t < 64) sB[t] = b1[t] + b2[t];
  __syncthreads();

  const int lane = t & 31;
  const int wave = t >> 5;
  const int half = lane >> 4;  // 0: lanes 0-15, 1: lanes 16-31
  const int n0 = lane & 15;

  const int gwave = blockIdx.x * 4 + wave;
  const int nwave = gridDim.x * 4;
  const int numTiles = N_NODES / 16;  // 5000, exact

  for (int tile = gwave; tile < numTiles; tile += nwave) {
    const int nb = tile * 16;
    v8f acc[4];
    for (int nt = 0; nt < 4; ++nt) {
      const float bv = sB[nt * 16 + n0];
      for (int v = 0; v < 8; ++v) acc[nt][v] = bv;
    }
    // h branch: K = 64
    const float* hrow = h + (size_t)(nb + n0) * 64;
    for (int kb = 0; kb < 64; kb += 4) {
      const v2f a = ld2_nt(hrow + kb + 2 * half);
      for (int nt = 0; nt < 4; ++nt)
        acc[nt] = wmma4(a, ld2(sW1 + (nt * 16 + n0) * 68 + kb + 2 * half), acc[nt]);
    }
    // e branch: K = 32
    const float* erow = e + (size_t)(nb + n0) * 32;
    for (int kb = 0; kb < 32; kb += 4) {
      const v2f a = ld2_nt(erow + kb + 2 * half);
      for (int nt = 0; nt < 4; ++nt)
        acc[nt] = wmma4(a, ld2(sW2 + (nt * 16 + n0) * 36 + kb + 2 * half), acc[nt]);
    }
    // store C layout: lane -> (m = half*8 + v, n = nt*16 + n0); 16-lane 64B chunks
    // p is immediately re-read (gathers) -> regular temporal stores keep it in L2.
    float* prow = p + (size_t)nb * 64;
    for (int nt = 0; nt < 4; ++nt)
      for (int v = 0; v < 8; ++v)
        prow[(half * 8 + v) * 64 + nt * 16 + n0] = acc[nt][v];
  }
}

// ---------------------------------------------------------------------------
// Kernel 2: per 16-edge tile (one tile per wave iteration):
//   s    = p[src] + p[dst] + e_ij @ W3^T + b3
//   hmid = elu(s @ Wh^T + bh)
//   out  = hmid @ Wo^T + bo
// s / hmid staged in a per-wave 16x68 LDS tile (conflict-free fragment reads).
// ---------------------------------------------------------------------------
__global__ __launch_bounds__(128) void edge_mlp_kernel(
    const float* __restrict__ e_ij, const int* __restrict__ src,
    const int* __restrict__ dst, const float* __restrict__ p,
    const float* __restrict__ W3, const float* __restrict__ b3,
    const float* __restrict__ Wh, const float* __restrict__ bh,
    const float* __restrict__ Wo, const float* __restrict__ bo,
    float* __restrict__ out) {
  __shared__ float sW3[64 * 20];      // W3[n][k], stride 20
  __shared__ float sWh[64 * 68];      // Wh[n][k], stride 68
  __shared__ float sWo[64 * 68];      // Wo[n][k], stride 68
  __shared__ float sBias[192];        // [b3 | bh | bo]
  __shared__ float sS[4][16 * 68];    // per-wave 16x64 tile, stride 68

  const int t = threadIdx.x;
  for (int i = t; i < 64 * 16; i += 128) sW3[(i >> 4) * 20 + (i & 15)] = W3[i];
  for (int i = t; i < 64 * 64; i += 128) {
    sWh[(i >> 6) * 68 + (i & 63)] = Wh[i];
    sWo[(i >> 6) * 68 + (i & 63)] = Wo[i];
  }
  if (t < 64) {
    sBias[t] = b3[t];
    sBias[64 + t] = bh[t];
    sBias[128 + t] = bo[t];
  }
  __syncthreads();

  const int lane = t & 31;
  const int wave = t >> 5;
  const int half = lane >> 4;
  const int n0 = lane & 15;
  float* sT = &sS[wave][0];

  const int gwave = blockIdx.x * 4 + wave;
  const int nwave = gridDim.x * 4;
  const int numTiles = N_EDGES / 16;  // 80000, exact

  for (int tile = gwave; tile < numTiles; tile += nwave) {
    const int eb = tile * 16;

    // ---- e5 = e_ij @ W3^T + b3 (K = 16) ----
    v8f acc[4];
    for (int nt = 0; nt < 4; ++nt) {
      const float bv = sBias[nt * 16 + n0];
      for (int v = 0; v < 8; ++v) acc[nt][v] = bv;
    }
    const float* arow = e_ij + (size_t)(eb + n0) * 16;
    for (int kb = 0; kb < 16; kb += 4) {
      const v2f a = ld2_nt(arow + kb + 2 * half);  // streamed once: NT
      for (int nt = 0; nt < 4; ++nt)
        acc[nt] = wmma4(a, ld2(sW3 + (nt * 16 + n0) * 20 + kb + 2 * half), acc[nt]);
    }
    // spill e5 (C layout) to LDS tile
    for (int nt = 0; nt < 4; ++nt)
      for (int v = 0; v < 8; ++v)
        sT[(half * 8 + v) * 68 + nt * 16 + n0] = acc[nt][v];
    lds_fence();

    // ---- gather-add: sT[m][:] += p[src[eb+m]][:] + p[dst[eb+m]][:] ----
    // half-wave reads one 256B p-row segment per step (coalesced, L2-resident)
    for (int j = 0; j < 8; ++j) {
      const int f = lane + 32 * j;      // 0..255
      const int m = f >> 4;             // edge row within tile
      const int c = (f & 15) * 4;       // float4 column
      const int si = src[eb + m];
      const int di = dst[eb + m];
      const float4 a4 = *(const float4*)(p + (size_t)si * 64 + c);
      const float4 b4 = *(const float4*)(p + (size_t)di * 64 + c);
      float4* sp = (float4*)(sT + m * 68 + c);
      float4 s4 = *sp;
      s4.x += a4.x + b4.x;
      s4.y += a4.y + b4.y;
      s4.z += a4.z + b4.z;
      s4.w += a4.w + b4.w;
      *sp = s4;
    }
    lds_fence();

    // ---- hmid = elu(s @ Wh^T + bh) (K = 64) ----
    v8f acc2[4];
    for (int nt = 0; nt < 4; ++nt) {
      const float bv = sBias[64 + nt * 16 + n0];
      for (int v = 0; v < 8; ++v) acc2[nt][v] = bv;
    }
    for (int kb = 0; kb < 64; kb += 4) {
      const v2f a = ld2(sT + n0 * 68 + kb + 2 * half);
      for (int nt = 0; nt < 4; ++nt)
        acc2[nt] = wmma4(a, ld2(sWh + (nt * 16 + n0) * 68 + kb + 2 * half), acc2[nt]);
    }
    lds_fence();  // all sT reads retired before overwrite
    for (int nt = 0; nt < 4; ++nt)
      for (int v = 0; v < 8; ++v) {
        const float x = acc2[nt][v];
        const float y = x > 0.0f ? x : (expf(x) - 1.0f);  // ELU, alpha=1
        sT[(half * 8 + v) * 68 + nt * 16 + n0] = y;
      }
    lds_fence();

    // ---- out = hmid @ Wo^T + bo (K = 64) ----
    v8f acc3[4];
    for (int nt = 0; nt < 4; ++nt) {
      const float bv = sBias[128 + nt * 16 + n0];
      for (int v = 0; v < 8; ++v) acc3[nt][v] = bv;
    }
    for (int kb = 0; kb < 64; kb += 4) {
      const v2f a = ld2(sT + n0 * 68 + kb + 2 * half);
      for (int nt = 0; nt < 4; ++nt)
        acc3[nt] = wmma4(a, ld2(sWo + (nt * 16 + n0) * 68 + kb + 2 * half), acc3[nt]);
    }
    // 328 MB streamed output: non-temporal stores, don't displace p in L2
    float* orow = out + (size_t)eb * 64;
    for (int nt = 0; nt < 4; ++nt)
      for (int v = 0; v < 8; ++v)
        __builtin_nontemporal_store(acc3[nt][v],
                                    orow + (half * 8 + v) * 64 + nt * 16 + n0);
  }
}

// ---------------------------------------------------------------------------
extern "C" void kernel_launch(void* const* d_in, const int* in_sizes, int n_in,
                              void* d_out, int out_size, void* d_ws,
                              size_t ws_size, hipStream_t stream) {
  const float* h    = (const float*)d_in[0];
  const float* e    = (const float*)d_in[1];
  const float* e_ij = (const float*)d_in[2];
  const int*   src  = (const int*)d_in[3];
  const int*   dst  = (const int*)d_in[4];
  const float* W1 = (const float*)d_in[5];
  const float* b1 = (const float*)d_in[6];
  const float* W2 = (const float*)d_in[7];
  const float* b2 = (const float*)d_in[8];
  const float* W3 = (const float*)d_in[9];
  const float* b3 = (const float*)d_in[10];
  const float* Wh = (const float*)d_in[11];
  const float* bh = (const float*)d_in[12];
  const float* Wo = (const float*)d_in[13];
  const float* bo = (const float*)d_in[14];
  float* out = (float*)d_out;
  float* p   = (float*)d_ws;  // [N_NODES x 64] node projection, 20.5 MB

  // 1250 blocks * 4 waves = 5000 waves -> exactly one 16-node tile each
  node_proj_kernel<<<1250, 128, 0, stream>>>(h, e, W1, b1, W2, b2, p);
  // 4000 blocks * 4 waves = 16000 waves -> 5 edge tiles each (80000 tiles)
  edge_mlp_kernel<<<4000, 128, 0, stream>>>(e_ij, src, dst, p, W3, b3, Wh, bh,
                                            Wo, bo, out);
}